// MeMDecoder_59450937311979
// MI455X (gfx1250) — compile-verified
//
#include <hip/hip_runtime.h>
#include <math.h>

#define BB    64
#define RNN   512
#define WVDIM 512
#define VOCAB 50000
#define SRCN  512
#define NMEM  512
#define OOV   50

typedef __attribute__((ext_vector_type(2))) float v2f;
typedef __attribute__((ext_vector_type(4))) float v4f;
typedef __attribute__((ext_vector_type(8))) float v8f;

// ---------------------------------------------------------------------------
// WMMA GEMM, C(64xN) = A(64xK) * W^T + bias, W stored row-major (N x K).
// One wave handles one 16-wide N tile and all four 16-row M tiles, so each
// b128 weight load feeds 4 independent v_wmma_f32_16x16x4_f32 chains.
// ---------------------------------------------------------------------------
template <bool NT>
__global__ void wmma_gemm_tn(const float* __restrict__ A,
                             const float* __restrict__ W,
                             const float* __restrict__ bias,
                             float* __restrict__ C, int N, int K) {
  const int lane   = threadIdx.x & 31;
  const int wave   = blockIdx.x * (blockDim.x >> 5) + (threadIdx.x >> 5);
  const int ntiles = N >> 4;
  if (wave >= ntiles) return;            // whole-wave uniform exit (EXEC stays all-1s)
  const int n0    = wave << 4;
  const int col   = lane & 15;           // N within tile / M within tile (A layout)
  const int hi    = lane >> 4;           // 0: K=k0,k0+1   1: K=k0+2,k0+3
  const int k_off = hi << 1;

  float bv = bias ? bias[n0 + col] : 0.0f;
  v8f acc0 = {bv, bv, bv, bv, bv, bv, bv, bv};
  v8f acc1 = acc0, acc2 = acc0, acc3 = acc0;

  const float* wrow = W + (size_t)(n0 + col) * K;   // one weight row per lane-column
  const float* a0p  = A + (size_t)(0 * 16 + col) * K + k_off;
  const float* a1p  = A + (size_t)(1 * 16 + col) * K + k_off;
  const float* a2p  = A + (size_t)(2 * 16 + col) * K + k_off;
  const float* a3p  = A + (size_t)(3 * 16 + col) * K + k_off;

  for (int k0 = 0; k0 < K; k0 += 4) {
    v4f wv;
    if (NT) wv = __builtin_nontemporal_load((const v4f*)(wrow + k0));
    else    wv = *(const v4f*)(wrow + k0);
    v2f b;
    b.x = hi ? wv.z : wv.x;              // B[k0+2*hi    ][n]
    b.y = hi ? wv.w : wv.y;              // B[k0+2*hi + 1][n]
    v2f a0 = *(const v2f*)(a0p + k0);
    v2f a1 = *(const v2f*)(a1p + k0);
    v2f a2 = *(const v2f*)(a2p + k0);
    v2f a3 = *(const v2f*)(a3p + k0);
    acc0 = __builtin_amdgcn_wmma_f32_16x16x4_f32(false, a0, false, b, (short)0, acc0, false, false);
    acc1 = __builtin_amdgcn_wmma_f32_16x16x4_f32(false, a1, false, b, (short)0, acc1, false, false);
    acc2 = __builtin_amdgcn_wmma_f32_16x16x4_f32(false, a2, false, b, (short)0, acc2, false, false);
    acc3 = __builtin_amdgcn_wmma_f32_16x16x4_f32(false, a3, false, b, (short)0, acc3, false, false);
  }

#pragma unroll
  for (int v = 0; v < 8; ++v) {
    int mr = v + (hi << 3);              // C layout: VGPR v -> M=v (lo) / v+8 (hi)
    C[(size_t)(0 * 16 + mr) * N + n0 + col] = acc0[v];
    C[(size_t)(1 * 16 + mr) * N + n0 + col] = acc1[v];
    C[(size_t)(2 * 16 + mr) * N + n0 + col] = acc2[v];
    C[(size_t)(3 * 16 + mr) * N + n0 + col] = acc3[v];
  }
}

// C(64xN) = A(64xK) * W, W stored row-major (K x N)   (q @ readq_w)
__global__ void wmma_gemm_nn(const float* __restrict__ A,
                             const float* __restrict__ W,
                             float* __restrict__ C, int N, int K) {
  const int lane   = threadIdx.x & 31;
  const int wave   = blockIdx.x * (blockDim.x >> 5) + (threadIdx.x >> 5);
  const int ntiles = N >> 4;
  if (wave >= ntiles) return;
  const int n0    = wave << 4;
  const int col   = lane & 15;
  const int hi    = lane >> 4;
  const int k_off = hi << 1;

  v8f acc0 = {0, 0, 0, 0, 0, 0, 0, 0};
  v8f acc1 = acc0, acc2 = acc0, acc3 = acc0;

  const float* a0p = A + (size_t)(0 * 16 + col) * K + k_off;
  const float* a1p = A + (size_t)(1 * 16 + col) * K + k_off;
  const float* a2p = A + (size_t)(2 * 16 + col) * K + k_off;
  const float* a3p = A + (size_t)(3 * 16 + col) * K + k_off;

  for (int k0 = 0; k0 < K; k0 += 4) {
    v2f b;
    b.x = W[(size_t)(k0 + k_off + 0) * N + n0 + col];
    b.y = W[(size_t)(k0 + k_off + 1) * N + n0 + col];
    v2f a0 = *(const v2f*)(a0p + k0);
    v2f a1 = *(const v2f*)(a1p + k0);
    v2f a2 = *(const v2f*)(a2p + k0);
    v2f a3 = *(const v2f*)(a3p + k0);
    acc0 = __builtin_amdgcn_wmma_f32_16x16x4_f32(false, a0, false, b, (short)0, acc0, false, false);
    acc1 = __builtin_amdgcn_wmma_f32_16x16x4_f32(false, a1, false, b, (short)0, acc1, false, false);
    acc2 = __builtin_amdgcn_wmma_f32_16x16x4_f32(false, a2, false, b, (short)0, acc2, false, false);
    acc3 = __builtin_amdgcn_wmma_f32_16x16x4_f32(false, a3, false, b, (short)0, acc3, false, false);
  }

#pragma unroll
  for (int v = 0; v < 8; ++v) {
    int mr = v + (hi << 3);
    C[(size_t)(0 * 16 + mr) * N + n0 + col] = acc0[v];
    C[(size_t)(1 * 16 + mr) * N + n0 + col] = acc1[v];
    C[(size_t)(2 * 16 + mr) * N + n0 + col] = acc2[v];
    C[(size_t)(3 * 16 + mr) * N + n0 + col] = acc3[v];
  }
}

// ---------------------------------------------------------------------------
// Small helper kernels
// ---------------------------------------------------------------------------
__global__ void k_build_xin(const float* __restrict__ c_t_1,
                            const float* __restrict__ emb,
                            const int* __restrict__ y,
                            float* __restrict__ xin) {
  int b = blockIdx.x;
  const float* erow = emb + (size_t)y[b] * WVDIM;
  for (int j = threadIdx.x; j < RNN; j += blockDim.x) {
    xin[(size_t)b * 1024 + j]       = c_t_1[(size_t)b * RNN + j];
    xin[(size_t)b * 1024 + RNN + j] = erow[j];
  }
}

__global__ void k_concat2(const float* __restrict__ h,
                          const float* __restrict__ ct,
                          float* __restrict__ xin) {
  int b = blockIdx.x;
  for (int j = threadIdx.x; j < RNN; j += blockDim.x) {
    xin[(size_t)b * 1024 + j]       = h[(size_t)b * RNN + j];
    xin[(size_t)b * 1024 + RNN + j] = ct[(size_t)b * RNN + j];
  }
}

__global__ void k_gru_gates(const float* __restrict__ gi,
                            const float* __restrict__ gh,
                            const float* __restrict__ h,
                            float* __restrict__ hout) {
  int idx = blockIdx.x * blockDim.x + threadIdx.x;
  if (idx >= BB * RNN) return;
  int b = idx / RNN, j = idx - b * RNN;
  const float* gib = gi + (size_t)b * 3 * RNN;
  const float* ghb = gh + (size_t)b * 3 * RNN;
  float r = 1.0f / (1.0f + __expf(-(gib[j] + ghb[j])));
  float z = 1.0f / (1.0f + __expf(-(gib[RNN + j] + ghb[RNN + j])));
  float n = tanhf(gib[2 * RNN + j] + r * ghb[2 * RNN + j]);
  hout[idx] = (1.0f - z) * n + z * h[idx];
}

// en[b,n] = (qw[b] . K[b,n]) * mask^2   (mask applied twice as in reference)
__global__ void k_attn_scores(const float* __restrict__ qw,
                              const float* __restrict__ Kmat,
                              const float* __restrict__ mask,
                              float* __restrict__ en, int Nn) {
  __shared__ float q[RNN];
  int b = blockIdx.x;
  for (int j = threadIdx.x; j < RNN; j += blockDim.x) q[j] = qw[(size_t)b * RNN + j];
  __syncthreads();
  for (int n = threadIdx.x; n < Nn; n += blockDim.x) {
    const float* krow = Kmat + ((size_t)b * Nn + n) * RNN;
    float s = 0.0f;
    for (int d = 0; d < RNN; d += 4) {
      v4f kv = *(const v4f*)(krow + d);
      s += q[d] * kv.x + q[d + 1] * kv.y + q[d + 2] * kv.z + q[d + 3] * kv.w;
    }
    float m = mask[(size_t)b * Nn + n];
    en[(size_t)b * Nn + n] = s * m * m;
  }
}

__device__ __forceinline__ float blockMax(float v, float* sm) {
  int lane = threadIdx.x & 31, wid = threadIdx.x >> 5, nw = blockDim.x >> 5;
#pragma unroll
  for (int o = 16; o > 0; o >>= 1) v = fmaxf(v, __shfl_down(v, o, 32));
  if (lane == 0) sm[wid] = v;
  __syncthreads();
  if (threadIdx.x == 0) {
    float m = sm[0];
    for (int i = 1; i < nw; ++i) m = fmaxf(m, sm[i]);
    sm[0] = m;
  }
  __syncthreads();
  float r = sm[0];
  __syncthreads();
  return r;
}

__device__ __forceinline__ float blockSum(float v, float* sm) {
  int lane = threadIdx.x & 31, wid = threadIdx.x >> 5, nw = blockDim.x >> 5;
#pragma unroll
  for (int o = 16; o > 0; o >>= 1) v += __shfl_down(v, o, 32);
  if (lane == 0) sm[wid] = v;
  __syncthreads();
  if (threadIdx.x == 0) {
    float s = sm[0];
    for (int i = 1; i < nw; ++i) s += sm[i];
    sm[0] = s;
  }
  __syncthreads();
  float r = sm[0];
  __syncthreads();
  return r;
}

// w = exp(en - max) * mask / (sum + 1e-10)
__global__ void k_attn_softmax(const float* __restrict__ en,
                               const float* __restrict__ mask,
                               float* __restrict__ w, int Nn) {
  __shared__ float sm[32];
  int b = blockIdx.x;
  const float* er = en + (size_t)b * Nn;
  const float* mr = mask + (size_t)b * Nn;
  float lmax = -1e30f;
  for (int n = threadIdx.x; n < Nn; n += blockDim.x) lmax = fmaxf(lmax, er[n]);
  lmax = blockMax(lmax, sm);
  float lsum = 0.0f;
  for (int n = threadIdx.x; n < Nn; n += blockDim.x) lsum += __expf(er[n] - lmax) * mr[n];
  lsum = blockSum(lsum, sm);
  float inv = 1.0f / (lsum + 1e-10f);
  for (int n = threadIdx.x; n < Nn; n += blockDim.x)
    w[(size_t)b * Nn + n] = __expf(er[n] - lmax) * mr[n] * inv;
}

// ctx[b,d] = sum_n w[b,n] * V[b,n,d]
__global__ void k_attn_ctx(const float* __restrict__ w,
                           const float* __restrict__ V,
                           float* __restrict__ ctx, int Nn) {
  __shared__ float ws[NMEM];
  int b = blockIdx.x;
  for (int n = threadIdx.x; n < Nn; n += blockDim.x) ws[n] = w[(size_t)b * Nn + n];
  __syncthreads();
  for (int d = threadIdx.x; d < RNN; d += blockDim.x) {
    float acc = 0.0f;
    const float* vb = V + (size_t)b * Nn * RNN + d;
    for (int n = 0; n < Nn; ++n) acc += ws[n] * vb[(size_t)n * RNN];
    ctx[(size_t)b * RNN + d] = acc;
  }
}

// final[b, 0:VOCAB] = 0.5 * softmax(logits[b]); final[b, VOCAB:VOCAB+OOV] = 0
__global__ void k_vocab_final(const float* __restrict__ logits,
                              float* __restrict__ fin) {
  __shared__ float sm[32];
  int b = blockIdx.x;
  const float* lr = logits + (size_t)b * VOCAB;
  float lmax = -1e30f;
  for (int v = threadIdx.x; v < VOCAB; v += blockDim.x) lmax = fmaxf(lmax, lr[v]);
  lmax = blockMax(lmax, sm);
  float lsum = 0.0f;
  for (int v = threadIdx.x; v < VOCAB; v += blockDim.x) lsum += __expf(lr[v] - lmax);
  lsum = blockSum(lsum, sm);
  float inv = 0.5f / lsum;
  float* fr = fin + (size_t)b * (VOCAB + OOV);
  for (int v = threadIdx.x; v < VOCAB; v += blockDim.x) fr[v] = __expf(lr[v] - lmax) * inv;
  for (int v = VOCAB + threadIdx.x; v < VOCAB + OOV; v += blockDim.x) fr[v] = 0.0f;
}

__global__ void k_scatter(const float* __restrict__ attn,
                          const int* __restrict__ idx,
                          float* __restrict__ fin) {
  int b = blockIdx.x, s = threadIdx.x;      // blockDim == SRCN
  atomicAdd(fin + (size_t)b * (VOCAB + OOV) + idx[(size_t)b * SRCN + s],
            0.5f * attn[(size_t)b * SRCN + s]);
}

// ---------------------------------------------------------------------------
extern "C" void kernel_launch(void* const* d_in, const int* in_sizes, int n_in,
                              void* d_out, int out_size, void* d_ws, size_t ws_size,
                              hipStream_t stream) {
  (void)in_sizes; (void)n_in; (void)out_size; (void)ws_size;
  const int*   y        = (const int*)d_in[0];
  const float* s_t_1    = (const float*)d_in[1];
  const float* enc_out  = (const float*)d_in[2];
  const float* enc_mask = (const float*)d_in[3];
  const float* c_t_1    = (const float*)d_in[4];
  const int*   ext_voc  = (const int*)d_in[6];
  const float* q_mem    = (const float*)d_in[7];
  const float* q_mask   = (const float*)d_in[8];
  const float* emb      = (const float*)d_in[9];
  const float* x_ctx_w  = (const float*)d_in[10];
  const float* x_ctx_b  = (const float*)d_in[11];
  const float* wih      = (const float*)d_in[12];
  const float* whh      = (const float*)d_in[13];
  const float* bih      = (const float*)d_in[14];
  const float* bhh      = (const float*)d_in[15];
  const float* readq_w  = (const float*)d_in[16];
  const float* readsrc_w= (const float*)d_in[17];
  const float* out1_w   = (const float*)d_in[18];
  const float* out1_b   = (const float*)d_in[19];
  const float* out2_w   = (const float*)d_in[20];
  const float* out2_b   = (const float*)d_in[21];

  // workspace layout (floats)
  float* ws = (float*)d_ws;
  size_t off = 0;
  float* logits = ws + off; off += (size_t)BB * VOCAB;   // 3.2 M
  float* xin    = ws + off; off += (size_t)BB * 1024;
  float* gi     = ws + off; off += (size_t)BB * 3 * RNN;
  float* gh     = ws + off; off += (size_t)BB * 3 * RNN;
  float* xbuf   = ws + off; off += (size_t)BB * RNN;     // x / o1
  float* h1     = ws + off; off += (size_t)BB * RNN;
  float* qw     = ws + off; off += (size_t)BB * RNN;
  float* en     = ws + off; off += (size_t)BB * RNN;
  float* aw     = ws + off; off += (size_t)BB * RNN;
  float* cmem   = ws + off; off += (size_t)BB * RNN;

  // output layout (floats): final | s_t | c_t | attn
  float* fin  = (float*)d_out;
  float* s_t  = fin + (size_t)BB * (VOCAB + OOV);
  float* c_t  = s_t + (size_t)BB * RNN;
  float* attn = c_t + (size_t)BB * RNN;

  const dim3 blk256(256);
  auto tn_blocks = [](int N) { return dim3(((N >> 4) + 7) / 8); };

  // 1) x = [c_t_1 | emb[y]] @ x_ctx_w^T + b
  k_build_xin<<<BB, blk256, 0, stream>>>(c_t_1, emb, y, xin);
  wmma_gemm_tn<false><<<tn_blocks(RNN), blk256, 0, stream>>>(xin, x_ctx_w, x_ctx_b, xbuf, RNN, 1024);

  // 2) GRU #1: h1 = gru(x, h0)
  wmma_gemm_tn<false><<<tn_blocks(3 * RNN), blk256, 0, stream>>>(xbuf, wih, bih, gi, 3 * RNN, RNN);
  wmma_gemm_tn<false><<<tn_blocks(3 * RNN), blk256, 0, stream>>>(s_t_1, whh, bhh, gh, 3 * RNN, RNN);
  k_gru_gates<<<(BB * RNN + 255) / 256, blk256, 0, stream>>>(gi, gh, s_t_1, h1);

  // 3) memory attention: c_mem
  wmma_gemm_nn<<<tn_blocks(RNN), blk256, 0, stream>>>(h1, readq_w, qw, RNN, RNN);
  k_attn_scores<<<BB, blk256, 0, stream>>>(qw, q_mem, q_mask, en, NMEM);
  k_attn_softmax<<<BB, blk256, 0, stream>>>(en, q_mask, aw, NMEM);
  k_attn_ctx<<<BB, blk256, 0, stream>>>(aw, q_mem, cmem, NMEM);

  // 4) GRU #2: h2 = gru(c_mem, h1)  -> straight into d_out s_t slot
  wmma_gemm_tn<false><<<tn_blocks(3 * RNN), blk256, 0, stream>>>(cmem, wih, bih, gi, 3 * RNN, RNN);
  wmma_gemm_tn<false><<<tn_blocks(3 * RNN), blk256, 0, stream>>>(h1, whh, bhh, gh, 3 * RNN, RNN);
  k_gru_gates<<<(BB * RNN + 255) / 256, blk256, 0, stream>>>(gi, gh, h1, s_t);

  // 5) source attention: c_t, attn_dist (both directly in d_out)
  wmma_gemm_nn<<<tn_blocks(RNN), blk256, 0, stream>>>(s_t, readsrc_w, qw, RNN, RNN);
  k_attn_scores<<<BB, blk256, 0, stream>>>(qw, enc_out, enc_mask, en, SRCN);
  k_attn_softmax<<<BB, blk256, 0, stream>>>(en, enc_mask, attn, SRCN);
  k_attn_ctx<<<BB, blk256, 0, stream>>>(attn, enc_out, c_t, SRCN);

  // 6) output projections: o1 = [h2|c_t] @ out1_w^T + b ; logits = o1 @ out2_w^T + b
  k_concat2<<<BB, blk256, 0, stream>>>(s_t, c_t, xin);
  wmma_gemm_tn<false><<<tn_blocks(RNN), blk256, 0, stream>>>(xin, out1_w, out1_b, xbuf, RNN, 1024);
  wmma_gemm_tn<true><<<tn_blocks(VOCAB), blk256, 0, stream>>>(xbuf, out2_w, out2_b, logits, VOCAB, RNN);

  // 7) final dist: 0.5*softmax, zero OOV tail (overwritten every call), scatter-add attention
  k_vocab_final<<<BB, dim3(512), 0, stream>>>(logits, fin);
  k_scatter<<<BB, dim3(SRCN), 0, stream>>>(attn, ext_voc, fin);
}